// SSTFRLayer_81149112090794
// MI455X (gfx1250) — compile-verified
//
#include <hip/hip_runtime.h>

// Problem constants (match reference)
#define SB 4
#define SL 64000
#define SD 128
#define CHUNK 256               // chunk length (multiple of 16)
#define NCHUNK (SL / CHUNK)     // 250
#define NTILE (CHUNK / 16)      // 16 WMMA time-tiles per chunk

typedef __attribute__((ext_vector_type(2))) float v2f;
typedef __attribute__((ext_vector_type(8))) float v8f;

struct cf { float r, i; };

__device__ __forceinline__ cf cmul(cf a, cf b) {
    return { a.r * b.r - a.i * b.i, a.r * b.i + a.i * b.r };
}
// h' = a*h + u  (complex fused)
__device__ __forceinline__ cf cfma(cf a, cf h, cf u) {
    return { fmaf(a.r, h.r, fmaf(-a.i, h.i, u.r)),
             fmaf(a.r, h.i, fmaf( a.i, h.r, u.i)) };
}

// Per-channel filter parameters: a = exp(-softplus(alpha_raw) + i*omega),
// a^-1, and b = exp(b_log_mag) * e^{i b_phase}
__device__ __forceinline__ void chan_params(const float* __restrict__ omega,
                                            const float* __restrict__ araw,
                                            const float* __restrict__ blm,
                                            const float* __restrict__ bph,
                                            int d, cf& a, cf& ainv, cf& bb) {
    float sp = log1pf(expf(araw[d]));      // softplus(alpha_raw) = -alpha
    float w  = omega[d];
    float sw, cw; sincosf(w, &sw, &cw);
    float ea = expf(-sp), eai = expf(sp);
    a    = { ea * cw,  ea * sw };
    ainv = { eai * cw, -eai * sw };
    float bm = expf(blm[d]);
    float sb, cb; sincosf(bph[d], &sb, &cb);
    bb = { bm * cb, bm * sb };
}

// ---------------- Pass 1: per-chunk local final state (h0 = 0) --------------
// Per 16-step tile:  h <- a^16 * h + a^15 * sum_s a^{-s} * b * x_s
// (inner sum is a plain dot product against a precomputed a^{-s}b table,
//  2 FMA/sample with no deep complex dependence chain)
// Workspace layout: partial[(b*SD + d)*NCHUNK + g]
__global__ __launch_bounds__(128) void sstfr_pass1(
    const float* __restrict__ x, const float* __restrict__ omega,
    const float* __restrict__ araw, const float* __restrict__ blm,
    const float* __restrict__ bph, float2* __restrict__ partial) {
    __shared__ float xs[CHUNK];
    int blk = blockIdx.x;
    int b = blk / NCHUNK, g = blk % NCHUNK;
    int d = threadIdx.x;
    long xbase = (long)b * SL + (long)g * CHUNK;
    xs[d]       = x[xbase + d];
    xs[d + 128] = x[xbase + d + 128];
    __syncthreads();

    cf a, ainv, bb;
    chan_params(omega, araw, blm, bph, d, a, ainv, bb);

    float wr[16], wi[16];
    cf a15, a16;
    {
        cf p = bb;
        #pragma unroll
        for (int s = 0; s < 16; ++s) { wr[s] = p.r; wi[s] = p.i; p = cmul(p, ainv); }
        cf a2 = cmul(a, a), a4 = cmul(a2, a2), a8 = cmul(a4, a4);
        a16 = cmul(a8, a8);
        a15 = cmul(cmul(a8, a4), cmul(a2, a));
    }
    cf h = {0.f, 0.f};
    for (int j = 0; j < NTILE; ++j) {
        cf acc = {0.f, 0.f};
        #pragma unroll
        for (int s = 0; s < 16; ++s) {
            float xv = xs[16*j + s];
            acc.r = fmaf(xv, wr[s], acc.r);
            acc.i = fmaf(xv, wi[s], acc.i);
        }
        h = cfma(a16, h, cmul(a15, acc));
    }
    partial[((long)b * SD + d) * NCHUNK + g] = make_float2(h.r, h.i);
}

// ---------------- Pass 2: log-depth scan over chunk summaries ---------------
// Constant chunk-to-chunk decay A = a^256 makes the recurrence a Hillis-Steele
// scan:  v_g += A^o * v_{g-o}, o = 1,2,4,...,128 (A^o by successive squaring).
// One block per (b,d); all traffic contiguous. Leaves the EXCLUSIVE carry
// c_g = A * S_{g-1} in place.
__global__ __launch_bounds__(256) void sstfr_pass2(
    const float* __restrict__ omega, const float* __restrict__ araw,
    const float* __restrict__ blm, const float* __restrict__ bph,
    float2* __restrict__ partial) {
    __shared__ float sr[256], si[256];
    int bd = blockIdx.x;                 // b*SD + d
    int d  = bd % SD;
    int g  = threadIdx.x;
    long base = (long)bd * NCHUNK;

    cf a, ainv, bb;
    chan_params(omega, araw, blm, bph, d, a, ainv, bb);
    cf A = a;                            // a^256 via 8 exact complex squarings
    #pragma unroll
    for (int i = 0; i < 8; ++i) A = cmul(A, A);

    cf cur = {0.f, 0.f};
    if (g < NCHUNK) { float2 u = partial[base + g]; cur = { u.x, u.y }; }

    cf Ap = A;
    #pragma unroll
    for (int o = 1; o < 256; o <<= 1) {
        sr[g] = cur.r; si[g] = cur.i;
        __syncthreads();
        if (g >= o) cur = cfma(Ap, cf{ sr[g - o], si[g - o] }, cur);
        __syncthreads();
        Ap = cmul(Ap, Ap);
    }
    // exclusive carry: c_g = A * S_{g-1}, c_0 = 0
    sr[g] = cur.r; si[g] = cur.i;
    __syncthreads();
    cf prev = (g > 0) ? cf{ sr[g - 1], si[g - 1] } : cf{ 0.f, 0.f };
    cf c = cmul(A, prev);
    if (g == 0) c = { 0.f, 0.f };
    if (g < NCHUNK) partial[base + g] = make_float2(c.r, c.i);
}

// ---------------- Pass 3: WMMA tiled scan + streaming output ----------------
// Within a 16-step tile: h_{t0+r} = a^r * (a*h_in + C_r),
//   C_r = sum_{s<=r} a^{-s} * b * x_{t0+s}  == (Tri16 @ V) with Tri shared
// Tri16 (lower-triangular ones, f32) is the A matrix of chained
// v_wmma_f32_16x16x4_f32; columns = 16 channels per wave; two matmuls per
// tile carry Re and Im of V so every lane owns both components of one channel.
__global__ __launch_bounds__(256) void sstfr_pass3(
    const float* __restrict__ x, const float* __restrict__ omega,
    const float* __restrict__ araw, const float* __restrict__ blm,
    const float* __restrict__ bph, const float2* __restrict__ carry_in,
    float2* __restrict__ out) {
    __shared__ float xs[CHUNK];
    int blk = blockIdx.x;
    int b = blk / NCHUNK, g = blk % NCHUNK;
    int tid = threadIdx.x;
    long xbase = (long)b * SL + (long)g * CHUNK;
    xs[tid] = x[xbase + tid];
    __syncthreads();

    int wave = tid >> 5;
    int lane = tid & 31;
    int n = lane & 15;
    bool lo = lane < 16;
    int c = (wave << 4) + n;                 // this lane's channel

    cf a, ainv, bb;
    chan_params(omega, araw, blm, bph, c, a, ainv, bb);

    // Lane's K-slots per 16x4 A/B register pair (documented 32-bit A layout:
    // VGPR0 holds K = lo?0:2, VGPR1 holds K = lo?1:3 within each 4-col chunk).
    int sA = lo ? 0 : 2, sB = lo ? 1 : 3;

    // w_s = a^{-s} * b  pre-scale table, selected into this lane's slots
    float wrA[4], wiA[4], wrB[4], wiB[4];
    {
        float wr[16], wi[16];
        cf p = bb;
        #pragma unroll
        for (int s = 0; s < 16; ++s) { wr[s] = p.r; wi[s] = p.i; p = cmul(p, ainv); }
        #pragma unroll
        for (int k = 0; k < 4; ++k) {
            wrA[k] = lo ? wr[4*k + 0] : wr[4*k + 2];
            wiA[k] = lo ? wi[4*k + 0] : wi[4*k + 2];
            wrB[k] = lo ? wr[4*k + 1] : wr[4*k + 3];
            wiB[k] = lo ? wi[4*k + 1] : wi[4*k + 3];
        }
    }
    // a^r post-scale for this lane's 8 row-slots (r = v + 8*hi)
    float rpr[8], rpi[8];
    {
        float pr[16], pi[16];
        cf p = {1.f, 0.f};
        #pragma unroll
        for (int s = 0; s < 16; ++s) { pr[s] = p.r; pi[s] = p.i; p = cmul(p, a); }
        #pragma unroll
        for (int v = 0; v < 8; ++v) {
            rpr[v] = lo ? pr[v] : pr[v + 8];
            rpi[v] = lo ? pi[v] : pi[v + 8];
        }
    }
    // Shared triangular A matrix chunks (constant per lane)
    v2f Amat[4];
    #pragma unroll
    for (int k = 0; k < 4; ++k) {
        Amat[k] = (v2f){ n >= 4*k + sA ? 1.f : 0.f,
                         n >= 4*k + sB ? 1.f : 0.f };
    }

    float2 cin = carry_in[((long)b * SD + c) * NCHUNK + g];
    cf h = { cin.x, cin.y };                 // state before this chunk

    long obase = ((long)b * SL + (long)g * CHUNK) * SD;

    for (int j = 0; j < NTILE; ++j) {
        float xA[4], xB[4];
        #pragma unroll
        for (int k = 0; k < 4; ++k) {
            xA[k] = xs[16*j + 4*k + sA];
            xB[k] = xs[16*j + 4*k + sB];
        }
        v8f cre = {0.f,0.f,0.f,0.f,0.f,0.f,0.f,0.f};
        v8f cim = {0.f,0.f,0.f,0.f,0.f,0.f,0.f,0.f};
        #pragma unroll
        for (int k = 0; k < 4; ++k) {
            v2f Bre = (v2f){ xA[k] * wrA[k], xB[k] * wrB[k] };
            v2f Bim = (v2f){ xA[k] * wiA[k], xB[k] * wiB[k] };
            cre = __builtin_amdgcn_wmma_f32_16x16x4_f32(
                false, Amat[k], false, Bre, (short)0, cre, false, false);
            cim = __builtin_amdgcn_wmma_f32_16x16x4_f32(
                false, Amat[k], false, Bim, (short)0, cim, false, false);
        }
        cf z = cmul(a, h);                   // a * h_in (all lanes replicated)
        float h7r = 0.f, h7i = 0.f;
        #pragma unroll
        for (int v = 0; v < 8; ++v) {
            cf q  = { cre[v] + z.r, cim[v] + z.i };
            cf hv = cmul(cf{rpr[v], rpi[v]}, q);
            int t = 16*j + v + (lo ? 0 : 8);
            long idx = obase + (long)t * SD + c;
            union { float2 f; unsigned long long u; } pun;
            pun.f = make_float2(hv.r, hv.i);
            __builtin_nontemporal_store(pun.u, (unsigned long long*)&out[idx]);
            if (v == 7) { h7r = hv.r; h7i = hv.i; }
        }
        // new carry = h at r=15, held by the hi half (v=7); broadcast per column
        int src = n + 16;
        h.r = __shfl(h7r, src, 32);
        h.i = __shfl(h7i, src, 32);
    }
}

extern "C" void kernel_launch(void* const* d_in, const int* in_sizes, int n_in,
                              void* d_out, int out_size, void* d_ws, size_t ws_size,
                              hipStream_t stream) {
    const float* x     = (const float*)d_in[0];
    const float* omega = (const float*)d_in[1];
    const float* araw  = (const float*)d_in[2];
    const float* blm   = (const float*)d_in[3];
    const float* bph   = (const float*)d_in[4];
    float2* partial = (float2*)d_ws;          // [b][d][g] complex, 1 MB
    float2* out     = (float2*)d_out;         // (B, L, D) complex64 interleaved

    sstfr_pass1<<<SB * NCHUNK, 128, 0, stream>>>(x, omega, araw, blm, bph, partial);
    sstfr_pass2<<<SB * SD, 256, 0, stream>>>(omega, araw, blm, bph, partial);
    sstfr_pass3<<<SB * NCHUNK, 256, 0, stream>>>(x, omega, araw, blm, bph, partial, out);
}